// Transformer_28905129902584
// MI455X (gfx1250) — compile-verified
//
#include <hip/hip_runtime.h>
#include <hip/hip_bf16.h>

typedef __attribute__((ext_vector_type(2))) float v2f;
typedef __attribute__((ext_vector_type(8))) float v8f;

#define SEQ   8192
#define VOCAB 10

// Single-instruction butterfly xor-shuffle via ds_swizzle (group-of-32 mode):
// offset[14:10] = xor_mask, offset[9:5] = or_mask(0), offset[4:0] = and_mask(0x1F).
// All 32 lanes active wherever this is used, so no bounds clamping is needed.
template<int XMASK>
__device__ __forceinline__ float swz_xor(float v) {
    return __int_as_float(
        __builtin_amdgcn_ds_swizzle(__float_as_int(v), (XMASK << 10) | 0x1F));
}

// Reductions across each 16-lane half of a wave32 (xor bits 0..3 never cross bit 4).
__device__ __forceinline__ float half16_max(float v) {
    v = fmaxf(v, swz_xor<1>(v));
    v = fmaxf(v, swz_xor<2>(v));
    v = fmaxf(v, swz_xor<4>(v));
    v = fmaxf(v, swz_xor<8>(v));
    return v;
}
__device__ __forceinline__ float half16_sum(float v) {
    v += swz_xor<1>(v);
    v += swz_xor<2>(v);
    v += swz_xor<4>(v);
    v += swz_xor<8>(v);
    return v;
}

// Kernel 1 (1 workgroup, 256 threads = 8 wave32):
//   A: LDS histogram of token ids (integer atomics -> deterministic)
//   B: wave 0 builds Q/K tables in WMMA A/B register layout, runs
//      V_WMMA_F32_16X16X4_F32 once, spills the 16x16 score tile to LDS
//   C: all 256 threads map 1:1 to (row, col) of the tile; count-weighted,
//      max-stabilized softmax per row via ds_swizzle butterflies (8 waves
//      in parallel, 2 rows per wave), emitting the 16x4 output table.
__global__ __launch_bounds__(256)
void attn_table_kernel(const int*   __restrict__ x,
                       const float* __restrict__ emb,   // [10,3]
                       const float* __restrict__ Wq, const float* __restrict__ bq,
                       const float* __restrict__ Wk, const float* __restrict__ bk,
                       const float* __restrict__ Wv, const float* __restrict__ bv,
                       float* __restrict__ Otab)        // [16,4] in d_ws
{
    __shared__ int   hist[16];
    __shared__ float S[16][16];
    const int tid = threadIdx.x;

    if (tid < 16) hist[tid] = 0;
    __syncthreads();
    for (int i = tid; i < SEQ; i += 256) {
        __builtin_prefetch(x + (i + 256 < SEQ ? i + 256 : i), 0, 1); // global_prefetch_b8
        atomicAdd(&hist[x[i]], 1);
    }
    __syncthreads();

    if (tid < 32) {                 // one full wave32, EXEC all-ones for WMMA
        const int  t     = tid & 15;   // A row / B col index
        const int  pr    = tid >> 4;   // which K-dim pair this half holds
        const bool valid = (t < VOCAB);

        float e0 = 0.f, e1 = 0.f, e2 = 0.f;
        if (valid) { e0 = emb[t*3+0]; e1 = emb[t*3+1]; e2 = emb[t*3+2]; }
        const int d0 = 2*pr, d1 = d0 + 1;

        // A = Q-table (16x4 f32): lanes 0-15 hold dims {0,1}, lanes 16-31 dims {2,3}
        // B = K^T     (4x16 f32): identical per-lane packing of row-major K-table
        v2f a, b;
        a.x = valid ? fmaf(e2, Wq[8+d0], fmaf(e1, Wq[4+d0], fmaf(e0, Wq[d0], bq[d0]))) : 0.f;
        a.y = valid ? fmaf(e2, Wq[8+d1], fmaf(e1, Wq[4+d1], fmaf(e0, Wq[d1], bq[d1]))) : 0.f;
        b.x = valid ? fmaf(e2, Wk[8+d0], fmaf(e1, Wk[4+d0], fmaf(e0, Wk[d0], bk[d0]))) : 0.f;
        b.y = valid ? fmaf(e2, Wk[8+d1], fmaf(e1, Wk[4+d1], fmaf(e0, Wk[d1], bk[d1]))) : 0.f;

        v8f c = {};
        c = __builtin_amdgcn_wmma_f32_16x16x4_f32(
                /*neg_a=*/false, a, /*neg_b=*/false, b,
                /*c_mod=*/(short)0, c, /*reuse_a=*/false, /*reuse_b=*/false);

        // D layout: vgpr r, lanes 0-15 -> row r, lanes 16-31 -> row r+8, col = t.
        #pragma unroll
        for (int r = 0; r < 8; ++r)
            S[r + 8*pr][t] = c[r];
    }
    __syncthreads();

    // Phase C: 256 threads == 16 rows x 16 cols; each wave32 = rows 2w, 2w+1.
    const int  row  = tid >> 4;
    const int  col  = tid & 15;
    const bool vcol = (col < VOCAB);

    float cnt = vcol ? (float)hist[col] : 0.f;
    float vv0 = 0.f, vv1 = 0.f, vv2 = 0.f, vv3 = 0.f;
    if (vcol) {
        const float e0 = emb[col*3+0], e1 = emb[col*3+1], e2 = emb[col*3+2];
        vv0 = fmaf(e2, Wv[8+0], fmaf(e1, Wv[4+0], fmaf(e0, Wv[0], bv[0])));
        vv1 = fmaf(e2, Wv[8+1], fmaf(e1, Wv[4+1], fmaf(e0, Wv[1], bv[1])));
        vv2 = fmaf(e2, Wv[8+2], fmaf(e1, Wv[4+2], fmaf(e0, Wv[2], bv[2])));
        vv3 = fmaf(e2, Wv[8+3], fmaf(e1, Wv[4+3], fmaf(e0, Wv[3], bv[3])));
    }

    const float s  = 0.5f * S[row][col];                 // 1/sqrt(D_HEAD) scaling
    float sm = (cnt > 0.f) ? s : -__builtin_inff();
    sm = half16_max(sm);                                 // row max (per 16-lane half)
    const float e  = (cnt > 0.f) ? cnt * __expf(s - sm) : 0.f;
    const float Z  = half16_sum(e);
    const float w0 = half16_sum(e * vv0);
    const float w1 = half16_sum(e * vv1);
    const float w2 = half16_sum(e * vv2);
    const float w3 = half16_sum(e * vv3);

    if (col == 0) {                                      // Z >= max-term count >= 1
        const float inv = 1.0f / Z;
        Otab[row*4+0] = w0 * inv;
        Otab[row*4+1] = w1 * inv;
        Otab[row*4+2] = w2 * inv;
        Otab[row*4+3] = w3 * inv;
    }
}

// Kernel 2: out[i] = O[x[i]], one 128-bit load + one 128-bit store per row.
__global__ __launch_bounds__(256)
void gather_kernel(const int* __restrict__ x,
                   const float4* __restrict__ Otab,
                   float4* __restrict__ out)
{
    const int i = blockIdx.x * 256 + threadIdx.x;
    if (i < SEQ) out[i] = Otab[x[i]];
}

extern "C" void kernel_launch(void* const* d_in, const int* in_sizes, int n_in,
                              void* d_out, int out_size, void* d_ws, size_t ws_size,
                              hipStream_t stream)
{
    const int*   x   = (const int*)  d_in[0];
    const float* emb = (const float*)d_in[1];
    const float* Wq  = (const float*)d_in[2];
    const float* bq  = (const float*)d_in[3];
    const float* Wk  = (const float*)d_in[4];
    const float* bk  = (const float*)d_in[5];
    const float* Wv  = (const float*)d_in[6];
    const float* bv  = (const float*)d_in[7];
    float* Otab = (float*)d_ws;   // 16x4 floats = 256 bytes of scratch

    attn_table_kernel<<<1, 256, 0, stream>>>(x, emb, Wq, bq, Wk, bk, Wv, bv, Otab);
    gather_kernel<<<SEQ / 256, 256, 0, stream>>>(x, (const float4*)Otab, (float4*)d_out);
}